// MaterialPointModel3d_38285338477003
// MI455X (gfx1250) — compile-verified
//
#include <hip/hip_runtime.h>
#include <hip/hip_bf16.h>
#include <stdint.h>

// ---------------- constants (match reference) ----------------
#define NGRID   128
#define G3      (NGRID * NGRID * NGRID)
#define DXc     (1.0f / 128.0f)
#define INV_DXc 128.0f
#define DTc     1.0e-4f
#define P_VOLc  ((DXc * 0.5f) * (DXc * 0.5f))
#define P_MASSc (P_VOLc * 1.0f)
#define MU0c    (5000.0f / (2.0f * 1.2f))             // E/(2(1+nu))
#define LAM0c   (5000.0f * 0.2f / (1.2f * 0.6f))      // E*nu/((1+nu)(1-2nu))
#define BLK     256
#define TILE_DW (BLK * 9)                              // dwords per staged F/C tile

typedef unsigned int v4u __attribute__((ext_vector_type(4)));
typedef int          v8i __attribute__((ext_vector_type(8)));
typedef int          v4i __attribute__((ext_vector_type(4)));

// ---------------- TDM descriptor builder (D# per CDNA5 ISA ch.8) ------------
// 2-D tile: dim0 = row of `tile0` dwords, dim1 = `tile1` rows, row stride
// `stride0` dwords. tensor_dim0/1 bound OOB (loads: zero-fill; stores: drop).
__device__ __forceinline__ void tdm_desc(v4u& g0, v8i& g1, const void* gaddr,
                                         unsigned lds_byte, unsigned td0,
                                         unsigned td1, unsigned tile0,
                                         unsigned tile1, unsigned stride0) {
  uint64_t ga = (uint64_t)(uintptr_t)gaddr;
  g0[0] = 1u;                                          // count=1, user mode
  g0[1] = lds_byte;                                    // lds_addr
  g0[2] = (unsigned)(ga & 0xFFFFFFFFu);                // global_addr lo
  g0[3] = ((unsigned)(ga >> 32) & 0x1FFFFFFu) | (2u << 30);  // hi | type=2
  g1[0] = (int)(2u << 16);                             // data_size=4B, mask=0
  g1[1] = (int)((td0 & 0xFFFFu) << 16);                // tensor_dim0[15:0]
  g1[2] = (int)(((td0 >> 16) & 0xFFFFu) | ((td1 & 0xFFFFu) << 16));
  g1[3] = (int)(((td1 >> 16) & 0xFFFFu) | ((tile0 & 0xFFFFu) << 16));
  g1[4] = (int)(tile1 & 0xFFFFu);                      // tile_dim1, tile_dim2=0
  g1[5] = (int)stride0;                                // tensor_dim0_stride lo
  g1[6] = 0;
  g1[7] = 0;
}

__device__ __forceinline__ void tdm_load_1d(const float* gsrc,
                                            unsigned lds_byte,
                                            unsigned n_valid_dwords,
                                            unsigned tile_dwords) {
  v4u g0; v8i g1;
  tdm_desc(g0, g1, gsrc, lds_byte, n_valid_dwords, 1u, tile_dwords, 1u,
           tile_dwords);
  v4i z4 = {0, 0, 0, 0};
  v8i z8 = {0, 0, 0, 0, 0, 0, 0, 0};
  __builtin_amdgcn_tensor_load_to_lds(g0, g1, z4, z4, z8, 0);
}

__device__ __forceinline__ void tdm_store_2d(float* gdst, unsigned lds_byte,
                                             unsigned row_dwords,
                                             unsigned rows_valid,
                                             unsigned rows_tile,
                                             unsigned stride_dwords) {
  v4u g0; v8i g1;
  tdm_desc(g0, g1, gdst, lds_byte, row_dwords, rows_valid, row_dwords,
           rows_tile, stride_dwords);
  v4i z4 = {0, 0, 0, 0};
  v8i z8 = {0, 0, 0, 0, 0, 0, 0, 0};
  __builtin_amdgcn_tensor_store_from_lds(g0, g1, z4, z4, z8, 0);
}

// ---------------- 3x3 SVD via cyclic Jacobi on F^T F ------------------------
__device__ __forceinline__ void jacobi_rot(float A[3][3], float V[3][3],
                                           int p, int q) {
  float apq = A[p][q];
  if (fabsf(apq) < 1.0e-12f) return;
  float theta = (A[q][q] - A[p][p]) / (2.0f * apq);
  float t = copysignf(1.0f, theta) / (fabsf(theta) + sqrtf(theta * theta + 1.0f));
  float c = rsqrtf(t * t + 1.0f);
  float s = t * c;
#pragma unroll
  for (int k = 0; k < 3; ++k) {                        // A <- A * J
    float akp = A[k][p], akq = A[k][q];
    A[k][p] = c * akp - s * akq;
    A[k][q] = s * akp + c * akq;
  }
#pragma unroll
  for (int k = 0; k < 3; ++k) {                        // A <- J^T * A
    float apk = A[p][k], aqk = A[q][k];
    A[p][k] = c * apk - s * aqk;
    A[q][k] = s * apk + c * aqk;
  }
#pragma unroll
  for (int k = 0; k < 3; ++k) {                        // V <- V * J
    float vkp = V[k][p], vkq = V[k][q];
    V[k][p] = c * vkp - s * vkq;
    V[k][q] = s * vkp + c * vkq;
  }
}

__device__ __forceinline__ void svd3(const float F[3][3], float U[3][3],
                                     float sig[3], float V[3][3]) {
  float A[3][3];
#pragma unroll
  for (int i = 0; i < 3; ++i)
#pragma unroll
    for (int j = 0; j < 3; ++j) {
      A[i][j] = F[0][i] * F[0][j] + F[1][i] * F[1][j] + F[2][i] * F[2][j];
      V[i][j] = (i == j) ? 1.0f : 0.0f;
    }
  for (int sweep = 0; sweep < 5; ++sweep) {
    jacobi_rot(A, V, 0, 1);
    jacobi_rot(A, V, 0, 2);
    jacobi_rot(A, V, 1, 2);
  }
  float lam[3] = {A[0][0], A[1][1], A[2][2]};
  // sort descending (match jnp.linalg.svd ordering), swap V columns along
#pragma unroll
  for (int a = 0; a < 2; ++a)
#pragma unroll
    for (int b = a + 1; b < 3; ++b) {
      if (lam[a] < lam[b]) {
        float t = lam[a]; lam[a] = lam[b]; lam[b] = t;
        for (int r = 0; r < 3; ++r) {
          float tv = V[r][a]; V[r][a] = V[r][b]; V[r][b] = tv;
        }
      }
    }
#pragma unroll
  for (int k = 0; k < 3; ++k) {
    sig[k] = sqrtf(fmaxf(lam[k], 0.0f));
    float ux = F[0][0] * V[0][k] + F[0][1] * V[1][k] + F[0][2] * V[2][k];
    float uy = F[1][0] * V[0][k] + F[1][1] * V[1][k] + F[1][2] * V[2][k];
    float uz = F[2][0] * V[0][k] + F[2][1] * V[1][k] + F[2][2] * V[2][k];
    float inv = 1.0f / fmaxf(sig[k], 1.0e-12f);
    U[0][k] = ux * inv; U[1][k] = uy * inv; U[2][k] = uz * inv;
  }
}

// weights / base helper (shared by P2G and G2P)
__device__ __forceinline__ void quad_weights(float px, float py, float pz,
                                             int base[3], float fx[3],
                                             float w[3][3]) {
  fx[0] = px * INV_DXc; fx[1] = py * INV_DXc; fx[2] = pz * INV_DXc;
#pragma unroll
  for (int d = 0; d < 3; ++d) {
    base[d] = (int)floorf(fx[d] - 0.5f);
    fx[d] = fx[d] - (float)base[d];
    w[0][d] = 0.5f * (1.5f - fx[d]) * (1.5f - fx[d]);
    w[1][d] = 0.75f - (fx[d] - 1.0f) * (fx[d] - 1.0f);
    w[2][d] = 0.5f * (fx[d] - 0.5f) * (fx[d] - 0.5f);
  }
}

// ---------------- kernels ---------------------------------------------------
__global__ void __launch_bounds__(256) k_zero_grid(float4* g) {
  int f = blockIdx.x * 256 + threadIdx.x;
  if (f < G3) g[f] = make_float4(0.f, 0.f, 0.f, 0.f);
}

__global__ void __launch_bounds__(BLK)
k_p2g(const float* __restrict__ x, const float* __restrict__ v,
      const float* __restrict__ Cin, const float* __restrict__ Fin,
      const float* __restrict__ Jpin, const int* __restrict__ material,
      float* __restrict__ grid4, float* __restrict__ out, int N) {
  __shared__ float ldsF[TILE_DW];
  __shared__ float ldsC[TILE_DW];
  __shared__ float ldsOut[BLK * 10];                   // Jp + Fn per particle
  int blockStart = blockIdx.x * BLK;
  unsigned rows_valid = (unsigned)(N - blockStart);
  // Kick off TDM loads of the F and C tiles first (wave 0; TDM ignores EXEC,
  // issues once per wave) so the DMA overlaps the x/v/weight computation.
  if ((threadIdx.x >> 5) == 0) {
    unsigned rem = rows_valid * 9u;
    tdm_load_1d(Fin + (size_t)blockStart * 9, (unsigned)(uintptr_t)&ldsF[0],
                rem, TILE_DW);
    tdm_load_1d(Cin + (size_t)blockStart * 9, (unsigned)(uintptr_t)&ldsC[0],
                rem, TILE_DW);
  }

  int p = blockStart + (int)threadIdx.x;
  bool active = p < N;
  int pc = active ? p : (N - 1);

  float px = x[pc * 3 + 0], py = x[pc * 3 + 1], pz = x[pc * 3 + 2];
  float vx = v[pc * 3 + 0], vy = v[pc * 3 + 1], vz = v[pc * 3 + 2];
  float jp = Jpin[pc];
  int mat = material[pc];
  int base[3]; float fx[3]; float w[3][3];
  quad_weights(px, py, pz, base, fx, w);

  if ((threadIdx.x >> 5) == 0) __builtin_amdgcn_s_wait_tensorcnt(0);
  __syncthreads();

  float Cm[3][3], Fm[3][3];
#pragma unroll
  for (int i = 0; i < 3; ++i)
#pragma unroll
    for (int j = 0; j < 3; ++j) {
      Cm[i][j] = ldsC[threadIdx.x * 9 + i * 3 + j];
      Fm[i][j] = ldsF[threadIdx.x * 9 + i * 3 + j];
    }

  // Fn = (I + dt*C) @ F
  float Fn[3][3];
#pragma unroll
  for (int i = 0; i < 3; ++i)
#pragma unroll
    for (int j = 0; j < 3; ++j)
      Fn[i][j] = Fm[i][j] + DTc * (Cm[i][0] * Fm[0][j] + Cm[i][1] * Fm[1][j] +
                                   Cm[i][2] * Fm[2][j]);

  float U[3][3], V[3][3], sig[3];
  svd3(Fn, U, sig, V);

  float h = fminf(fmaxf(expf(10.0f * (1.0f - jp)), 0.1f), 5.0f);
  if (mat == 1) h = 0.3f;
  float mu = (mat == 0) ? 0.0f : MU0c * h;
  float la = LAM0c * h;

  float J = 1.0f;  // reference multiplies only the first two sigmas
#pragma unroll
  for (int d = 0; d < 2; ++d) {
    float s0 = sig[d];
    float ns = (mat == 2) ? fminf(fmaxf(s0, 1.0f - 0.025f), 1.0f + 0.0045f) : s0;
    jp = jp * s0 / ns;
    sig[d] = ns;
    J *= ns;
  }

  float Fsnow[3][3], R[3][3];
#pragma unroll
  for (int i = 0; i < 3; ++i)
#pragma unroll
    for (int j = 0; j < 3; ++j) {
      Fsnow[i][j] = U[i][0] * sig[0] * V[j][0] + U[i][1] * sig[1] * V[j][1] +
                    U[i][2] * sig[2] * V[j][2];
      R[i][j] = U[i][0] * V[j][0] + U[i][1] * V[j][1] + U[i][2] * V[j][2];
    }

  float sqJ = sqrtf(J);
#pragma unroll
  for (int i = 0; i < 3; ++i)
#pragma unroll
    for (int j = 0; j < 3; ++j) {
      float f = Fn[i][j];
      if (mat == 0) f = (i == j) ? sqJ : 0.0f;
      else if (mat == 2) f = Fsnow[i][j];
      Fn[i][j] = f;
    }

  // stress = coef*(2 mu (Fn-R) Fn^T + I la J (J-1)); affine = stress + m*C
  const float coef = -DTc * P_VOLc * 4.0f * INV_DXc * INV_DXc;
  float lj = la * J * (J - 1.0f);
  float aff[3][3];
#pragma unroll
  for (int i = 0; i < 3; ++i)
#pragma unroll
    for (int j = 0; j < 3; ++j) {
      float acc = (Fn[i][0] - R[i][0]) * Fn[j][0] +
                  (Fn[i][1] - R[i][1]) * Fn[j][1] +
                  (Fn[i][2] - R[i][2]) * Fn[j][2];
      float st = coef * (2.0f * mu * acc + ((i == j) ? lj : 0.0f));
      aff[i][j] = st + P_MASSc * Cm[i][j];
    }

  // stage Jp + Fn row (out cols 6..15) in LDS; TDM scatters it at the end
  ldsOut[threadIdx.x * 10 + 0] = jp;
#pragma unroll
  for (int i = 0; i < 3; ++i)
#pragma unroll
    for (int j = 0; j < 3; ++j)
      ldsOut[threadIdx.x * 10 + 1 + i * 3 + j] = Fn[i][j];

  if (active) {
#pragma unroll
    for (int i = 0; i < 3; ++i) {
#pragma unroll
      for (int j = 0; j < 3; ++j) {
        float wij = w[i][0] * w[j][1];  // reference omits the z-dim factor
        float dpx = ((float)i - fx[0]) * DXc;
        float dpy = ((float)j - fx[1]) * DXc;
#pragma unroll
        for (int k = 0; k < 3; ++k) {
          float dpz = ((float)k - fx[2]) * DXc;
          int cx = base[0] + i, cy = base[1] + j, cz = base[2] + k;
          if (cx >= 0 && cx < NGRID && cy >= 0 && cy < NGRID && cz >= 0 &&
              cz < NGRID) {
            int f = (cx * NGRID + cy) * NGRID + cz;
            float mx = wij * (P_MASSc * vx + aff[0][0] * dpx + aff[0][1] * dpy + aff[0][2] * dpz);
            float my = wij * (P_MASSc * vy + aff[1][0] * dpx + aff[1][1] * dpy + aff[1][2] * dpz);
            float mz = wij * (P_MASSc * vz + aff[2][0] * dpx + aff[2][1] * dpy + aff[2][2] * dpz);
            atomicAdd(&grid4[f * 4 + 0], mx);
            atomicAdd(&grid4[f * 4 + 1], my);
            atomicAdd(&grid4[f * 4 + 2], mz);
            atomicAdd(&grid4[f * 4 + 3], wij * P_MASSc);
          }
        }
      }
    }
  }

  __syncthreads();  // all ldsOut rows written
  if ((threadIdx.x >> 5) == 0) {
    // strided 2D store: 10-dword rows into 25-dword output rows; rows beyond
    // rows_valid are dropped by tensor_dim1 OOB. S_ENDPGM drains TENSORcnt.
    tdm_store_2d(out + (size_t)blockStart * 25 + 6,
                 (unsigned)(uintptr_t)&ldsOut[0], 10u, rows_valid, BLK, 25u);
  }
}

__global__ void __launch_bounds__(256)
k_grid(float4* __restrict__ grid4, const float* __restrict__ gravity,
       const float* __restrict__ astr, const float* __restrict__ apos) {
  int f = blockIdx.x * 256 + threadIdx.x;
  if (f >= G3) return;
  float4 cell = grid4[f];
  float m = cell.w;
  bool mask = m > 0.0f;
  float inv = 1.0f / (mask ? m : 1.0f);
  float gx = cell.x * inv;
  float gy = cell.y * inv;
  float gz = cell.z * inv;
  gx += DTc * gravity[0] * 30.0f;
  gy += DTc * gravity[1] * 30.0f;
  gz += DTc * gravity[2] * 30.0f;
  int i = f >> 14, j = (f >> 7) & 127, k = f & 127;
  float dx0 = apos[0] - DXc * (float)i;
  float dy0 = apos[1] - DXc * (float)j;
  float dz0 = apos[2] - DXc * (float)k;
  float nrm = sqrtf(dx0 * dx0 + dy0 * dy0 + dz0 * dz0);
  float sc = astr[0] * DTc * 100.0f / (0.01f + nrm);
  gx += dx0 * sc;
  gy += dy0 * sc;
  gz += dz0 * sc;
  if (i < 3 && gx < 0.0f) gx = 0.0f;
  if (i > NGRID - 3 && gx > 0.0f) gx = 0.0f;
  if (j < 3 && gy < 0.0f) gy = 0.0f;
  if (j > NGRID - 3 && gy > 0.0f) gy = 0.0f;
  // reference applies no z boundary
  if (!mask) { gx = 0.0f; gy = 0.0f; gz = 0.0f; }
  grid4[f] = make_float4(gx, gy, gz, m);
}

__global__ void __launch_bounds__(BLK)
k_g2p(const float* __restrict__ x, const float4* __restrict__ grid4,
      float* __restrict__ out, int N) {
  __shared__ float ldsA[BLK * 6];                      // x_new + new_v
  __shared__ float ldsB[BLK * 9];                      // new_C
  int blockStart = blockIdx.x * BLK;
  unsigned rows_valid = (unsigned)(N - blockStart);
  int p = blockStart + (int)threadIdx.x;
  int pc = (p < N) ? p : (N - 1);

  float px = x[pc * 3 + 0], py = x[pc * 3 + 1], pz = x[pc * 3 + 2];
  int base[3]; float fx[3]; float w[3][3];
  quad_weights(px, py, pz, base, fx, w);

  float nv[3] = {0.f, 0.f, 0.f};
  float nC[3][3] = {{0.f, 0.f, 0.f}, {0.f, 0.f, 0.f}, {0.f, 0.f, 0.f}};
#pragma unroll
  for (int i = 0; i < 3; ++i) {
#pragma unroll
    for (int j = 0; j < 3; ++j) {
      float wij = w[i][0] * w[j][1];  // matches reference weighting
      float dpx = (float)i - fx[0];
      float dpy = (float)j - fx[1];
#pragma unroll
      for (int k = 0; k < 3; ++k) {
        float dpz = (float)k - fx[2];
        int cx = min(max(base[0] + i, 0), NGRID - 1);
        int cy = min(max(base[1] + j, 0), NGRID - 1);
        int cz = min(max(base[2] + k, 0), NGRID - 1);
        float4 gv = grid4[(cx * NGRID + cy) * NGRID + cz];  // one b128 load
        nv[0] += wij * gv.x;
        nv[1] += wij * gv.y;
        nv[2] += wij * gv.z;
        nC[0][0] += wij * gv.x * dpx; nC[0][1] += wij * gv.x * dpy; nC[0][2] += wij * gv.x * dpz;
        nC[1][0] += wij * gv.y * dpx; nC[1][1] += wij * gv.y * dpy; nC[1][2] += wij * gv.y * dpz;
        nC[2][0] += wij * gv.z * dpx; nC[2][1] += wij * gv.z * dpy; nC[2][2] += wij * gv.z * dpz;
      }
    }
  }
  ldsA[threadIdx.x * 6 + 0] = px + DTc * nv[0];
  ldsA[threadIdx.x * 6 + 1] = py + DTc * nv[1];
  ldsA[threadIdx.x * 6 + 2] = pz + DTc * nv[2];
  ldsA[threadIdx.x * 6 + 3] = nv[0];
  ldsA[threadIdx.x * 6 + 4] = nv[1];
  ldsA[threadIdx.x * 6 + 5] = nv[2];
#pragma unroll
  for (int c = 0; c < 3; ++c)
#pragma unroll
    for (int d = 0; d < 3; ++d)
      ldsB[threadIdx.x * 9 + c * 3 + d] = 4.0f * INV_DXc * nC[c][d];

  __syncthreads();
  if ((threadIdx.x >> 5) == 0) {
    tdm_store_2d(out + (size_t)blockStart * 25 + 0,
                 (unsigned)(uintptr_t)&ldsA[0], 6u, rows_valid, BLK, 25u);
    tdm_store_2d(out + (size_t)blockStart * 25 + 16,
                 (unsigned)(uintptr_t)&ldsB[0], 9u, rows_valid, BLK, 25u);
  }
}

// ---------------- launcher ---------------------------------------------------
extern "C" void kernel_launch(void* const* d_in, const int* in_sizes, int n_in,
                              void* d_out, int out_size, void* d_ws,
                              size_t ws_size, hipStream_t stream) {
  const float* x        = (const float*)d_in[0];
  const float* v        = (const float*)d_in[1];
  const float* C        = (const float*)d_in[2];
  const float* F        = (const float*)d_in[3];
  const float* Jp       = (const float*)d_in[4];
  const float* gravity  = (const float*)d_in[5];
  const float* astr     = (const float*)d_in[6];
  const float* apos     = (const float*)d_in[7];
  const int*   material = (const int*)d_in[8];
  int N = in_sizes[0] / 3;
  float* grid4 = (float*)d_ws;  // G3 * 4 floats = 33.5 MB (fits 192 MB L2)
  float* out = (float*)d_out;

  k_zero_grid<<<G3 / 256, 256, 0, stream>>>((float4*)grid4);
  int pb = (N + BLK - 1) / BLK;
  k_p2g<<<pb, BLK, 0, stream>>>(x, v, C, F, Jp, material, grid4, out, N);
  k_grid<<<G3 / 256, 256, 0, stream>>>((float4*)grid4, gravity, astr, apos);
  k_g2p<<<pb, BLK, 0, stream>>>(x, (const float4*)grid4, out, N);
}